// SparseMOEConv_35845797053215
// MI455X (gfx1250) — compile-verified
//
#include <hip/hip_runtime.h>
#include <math.h>

typedef float v2f __attribute__((ext_vector_type(2)));
typedef float v8f __attribute__((ext_vector_type(8)));

#define Bn 8
#define CIN 16
#define Hh 128
#define Wh 128
#define HW (Hh * Wh)            // 16384
#define NPIX (Bn * HW)          // 131072
#define NEXP 16
#define COUT 64
#define KDIM 144                // Cin * 3 * 3
#define TILE_N 64
#define MAXTILES ((2 * NPIX) / TILE_N + NEXP)   // 4112
#define LDA 148                 // padded LDS row stride (floats), conflict-free

// ---- workspace layout (bytes) ----
#define WS_COUNTS   0                       // 16 ints
#define WS_OFFSETS  64                      // 17 ints
#define WS_TILEBASE 192                     // 17 ints
#define WS_CURSORS  320                     // 16 ints
#define WS_ROUTE    512                     // NPIX * int4  (2 MB)
#define WS_LIST     (512 + NPIX * 16)       // 2*NPIX * int2 (2 MB)

// Native fp32 scatter-add: lowers to GLOBAL_ATOMIC_ADD_F32 (no CAS loop).
__device__ __forceinline__ void atomic_add_f32(float* p, float v) {
    __hip_atomic_fetch_add(p, v, __ATOMIC_RELAXED, __HIP_MEMORY_SCOPE_AGENT);
}

// ---------------- kernel 1: zero output region + counters ----------------
__global__ void moe_zero_kernel(float* __restrict__ out, int* __restrict__ counts) {
    int i = blockIdx.x * 256 + threadIdx.x;
    ((float4*)out)[i] = make_float4(0.f, 0.f, 0.f, 0.f);
    if (blockIdx.x == 0 && threadIdx.x < NEXP) counts[threadIdx.x] = 0;
}

// ---------------- kernel 2: gate (logits, top-2 routing) ----------------
__global__ void moe_gate_kernel(const float* __restrict__ x,
                                const float* __restrict__ gw,
                                const float* __restrict__ gb,
                                float* __restrict__ logits_out,
                                int4* __restrict__ route,
                                int* __restrict__ counts) {
    __shared__ float sW[16 * 16];
    __shared__ float sB[16];
    int tid = threadIdx.x;
    sW[tid & 255] = gw[tid & 255];
    if (tid < 16) sB[tid] = gb[tid];
    __syncthreads();

    int p = blockIdx.x * 256 + tid;
    int b = p >> 14;
    int hw = p & (HW - 1);
    const float* xp = x + ((size_t)b << 18) + hw;   // b * Cin * HW
    float xv[16];
#pragma unroll
    for (int ci = 0; ci < 16; ++ci) xv[ci] = xp[(size_t)ci << 14];

    float lg[16];
#pragma unroll
    for (int co = 0; co < 16; ++co) {
        float a = sB[co];
#pragma unroll
        for (int ci = 0; ci < 16; ++ci) a = fmaf(sW[co * 16 + ci], xv[ci], a);
        lg[co] = a;
        logits_out[(((size_t)b * 16 + co) << 14) + hw] = a;
    }
    // top-2 of logits (== top-2 of softmax); first-index tie-break like top_k
    float l0 = -INFINITY; int e0 = 0;
#pragma unroll
    for (int co = 0; co < 16; ++co) if (lg[co] > l0) { l0 = lg[co]; e0 = co; }
    float l1 = -INFINITY; int e1 = 0;
#pragma unroll
    for (int co = 0; co < 16; ++co) if (co != e0 && lg[co] > l1) { l1 = lg[co]; e1 = co; }
    float r = expf(l1 - l0);
    float inv = 1.f / (1.f + r);
    float w0 = inv, w1 = r * inv;

    route[p] = make_int4(e0, e1, __float_as_int(w0), __float_as_int(w1));
    atomicAdd(&counts[e0], 1);
    atomicAdd(&counts[e1], 1);
}

// ---------------- kernel 3: tiny exclusive scan ----------------
__global__ void moe_scan_kernel(const int* __restrict__ counts, int* __restrict__ offsets,
                                int* __restrict__ tileBase, int* __restrict__ cursors) {
    if (threadIdx.x == 0) {
        int o = 0, t = 0;
        for (int e = 0; e < NEXP; ++e) {
            offsets[e] = o; tileBase[e] = t; cursors[e] = o;
            o += counts[e];
            t += (counts[e] + TILE_N - 1) / TILE_N;
        }
        offsets[NEXP] = o; tileBase[NEXP] = t;
    }
}

// ---------------- kernel 4: compact per-expert pixel lists ----------------
__global__ void moe_fill_kernel(const int4* __restrict__ route, int* __restrict__ cursors,
                                int2* __restrict__ list) {
    int p = blockIdx.x * 256 + threadIdx.x;
    int4 r = route[p];
    int pos0 = atomicAdd(&cursors[r.x], 1);
    list[pos0] = make_int2(p, r.z);
    int pos1 = atomicAdd(&cursors[r.y], 1);
    list[pos1] = make_int2(p, r.w);
}

// ---------------- kernel 5: routed implicit-GEMM conv via fp32 WMMA ----------------
__global__ __launch_bounds__(256)
void moe_expert_kernel(const float* __restrict__ x, const float* __restrict__ ew,
                       const float* __restrict__ eb, float* __restrict__ out,
                       const int* __restrict__ tileBase, const int* __restrict__ offsets,
                       const int* __restrict__ counts, const int2* __restrict__ list) {
    __shared__ __align__(16) float As[COUT * LDA];     // weights: [co][k]
    __shared__ __align__(16) float Bs[TILE_N * LDA];   // patches transposed: [n][k]
    __shared__ int   sPid[TILE_N];
    __shared__ float sWt[TILE_N];
    __shared__ float sBias[COUT];

    int t = blockIdx.x;
    if (t >= tileBase[NEXP]) return;
    int e = 0;
    while (tileBase[e + 1] <= t) ++e;
    int tileInE = t - tileBase[e];
    int startIdx = offsets[e] + tileInE * TILE_N;
    int endIdx = offsets[e] + counts[e];
    int tid = threadIdx.x;

    if (tid < TILE_N) {
        int gi = startIdx + tid;
        if (gi < endIdx) { int2 en = list[gi]; sPid[tid] = en.x; sWt[tid] = __int_as_float(en.y); }
        else             { sPid[tid] = -1;     sWt[tid] = 0.f; }
    } else if (tid < TILE_N + COUT) {
        sBias[tid - TILE_N] = eb[e * COUT + (tid - TILE_N)];
    }
    // A: expert weights 64x144, flat co-major in memory == [co][ci*9+kh*3+kw]
    for (int i = tid; i < COUT * KDIM; i += 256) {
        int co = i / KDIM, k = i - co * KDIM;
        As[co * LDA + k] = ew[e * COUT * KDIM + i];
    }
    __syncthreads();
    // B: im2col gather of up to 64 pixel patches (zero-padded 3x3)
    for (int i = tid; i < TILE_N * KDIM; i += 256) {
        int n = i / KDIM, k = i - n * KDIM;
        float v = 0.f;
        int p = sPid[n];
        if (p >= 0) {
            int b = p >> 14, hw = p & (HW - 1);
            int h = hw >> 7, w = hw & (Wh - 1);
            int ci = k / 9, r9 = k - ci * 9;
            int kh = r9 / 3, kw = r9 - kh * 3;
            int hh = h + kh - 1, ww = w + kw - 1;
            if (hh >= 0 && hh < Hh && ww >= 0 && ww < Wh)
                v = x[(((size_t)b * CIN + ci) << 14) + (hh << 7) + ww];
        }
        Bs[n * LDA + k] = v;
    }
    __syncthreads();

    int wv = tid >> 5, lane = tid & 31;
    int row = lane & 15;            // A row (co) / B column (n) within tile
    int ksub = (lane >> 4) * 2;     // K sub-offset for upper half-wave
    for (int tt = wv; tt < 16; tt += 8) {     // 8 waves x 2 C-tiles = 4 Mtiles x 4 Ntiles
        int mt = tt & 3, nt = tt >> 2;
        const float* Ap = &As[(mt * 16 + row) * LDA + ksub];
        const float* Bp = &Bs[(nt * 16 + row) * LDA + ksub];
        v8f c = {};
#pragma unroll
        for (int k0 = 0; k0 < KDIM; k0 += 4) {
            v2f a = *(const v2f*)(Ap + k0);
            v2f bfrag = *(const v2f*)(Bp + k0);
            c = __builtin_amdgcn_wmma_f32_16x16x4_f32(false, a, false, bfrag,
                                                      (short)0, c, false, false);
        }
        int n = nt * 16 + (lane & 15);
        int p = sPid[n];
        if (p >= 0) {
            float wgt = sWt[n];
            int b = p >> 14, hw = p & (HW - 1);
            int half8 = (lane >> 4) * 8;
            float* op = out + (((size_t)b * COUT) << 14) + hw;
#pragma unroll
            for (int r = 0; r < 8; ++r) {
                int m = mt * 16 + r + half8;
                atomic_add_f32(op + ((size_t)m << 14), wgt * (c[r] + sBias[m]));
            }
        }
    }
}

extern "C" void kernel_launch(void* const* d_in, const int* in_sizes, int n_in,
                              void* d_out, int out_size, void* d_ws, size_t ws_size,
                              hipStream_t stream) {
    const float* x  = (const float*)d_in[0];
    const float* gw = (const float*)d_in[1];
    const float* gb = (const float*)d_in[2];
    const float* ew = (const float*)d_in[3];
    const float* eb = (const float*)d_in[4];
    float* out    = (float*)d_out;
    float* logits = out + (size_t)Bn * COUT * HW;

    char* ws = (char*)d_ws;
    int*  counts   = (int*)(ws + WS_COUNTS);
    int*  offsets  = (int*)(ws + WS_OFFSETS);
    int*  tileBase = (int*)(ws + WS_TILEBASE);
    int*  cursors  = (int*)(ws + WS_CURSORS);
    int4* route    = (int4*)(ws + WS_ROUTE);
    int2* list     = (int2*)(ws + WS_LIST);

    moe_zero_kernel<<<(Bn * COUT * HW / 4) / 256, 256, 0, stream>>>(out, counts);
    moe_gate_kernel<<<NPIX / 256, 256, 0, stream>>>(x, gw, gb, logits, route, counts);
    moe_scan_kernel<<<1, 32, 0, stream>>>(counts, offsets, tileBase, cursors);
    moe_fill_kernel<<<NPIX / 256, 256, 0, stream>>>(route, cursors, list);
    moe_expert_kernel<<<MAXTILES, 256, 0, stream>>>(x, ew, eb, out,
                                                    tileBase, offsets, counts, list);
}